// Gaussian2DImage_20160576487776
// MI455X (gfx1250) — compile-verified
//
#include <hip/hip_runtime.h>

typedef __attribute__((ext_vector_type(2))) float v2f;
typedef __attribute__((ext_vector_type(8))) float v8f;

#define NG      1024
#define JITTER  1e-6f
#define TWO_PI  6.28318530717958647692f
#define KEXP    (-0.72134752044448170368f)   // -0.5 * log2(e)

// ---------------------------------------------------------------------------
// Kernel 1: per-gaussian activation + inverse-covariance -> 8 quadratic coefs,
// pre-scaled by -0.5*log2(e) so the WMMA output is directly the exp2 argument:
//   alpha = op * exp(-q/2) = exp2( A'x^2 + B'xy + C'y^2 + D'x + E'y + F' )
//
// Store permutation for merge-free compositing: gaussian g goes to A-matrix
// slot  (g<512)  ->  tile g/8,      row g%8        (lanes 0-15 of D)
//       (g>=512) ->  tile (g-512)/8, row 8+(g-512)%8 (lanes 16-31 of D)
// so the lo lane-half composites the ordered segment [0,512) and the hi half
// [512,1024); one final xor-16 merge reconstructs the full ordered product.
// ---------------------------------------------------------------------------
__global__ void g2d_precompute(const float* __restrict__ means,
                               const float* __restrict__ scales,
                               const float* __restrict__ thetas,
                               const float* __restrict__ opac,
                               float* __restrict__ coefs /* [NG][8] */) {
    int g = blockIdx.x * blockDim.x + threadIdx.x;
    if (g >= NG) return;

    float mx = tanhf(means[2 * g + 0]);
    float my = tanhf(means[2 * g + 1]);
    float s0 = expf(scales[2 * g + 0]);
    float s1 = expf(scales[2 * g + 1]);
    float th = TWO_PI / (1.0f + expf(-thetas[g]));
    float op = 1.0f / (1.0f + expf(-opac[g]));

    float si, c;
    __sincosf(th, &si, &c);
    float s02 = s0 * s0, s12 = s1 * s1;
    float a = c * c * s02 + si * si * s12 + JITTER;   // Sigma_00
    float b = c * si * (s02 - s12);                   // Sigma_01
    float d = si * si * s02 + c * c * s12 + JITTER;   // Sigma_11
    float inv = 1.0f / (a * d - b * b);
    float ia = d * inv, ib = -b * inv, idd = a * inv; // Sigma^{-1}

    float A = ia;
    float B = 2.0f * ib;
    float C = idd;
    float D = -2.0f * (ia * mx + ib * my);
    float E = -2.0f * (ib * mx + idd * my);
    float F = ia * mx * mx + 2.0f * ib * mx * my + idd * my * my
              - 2.0f * logf(op);                      // fold opacity in

    // Segment-permuted slot (see header comment).
    int seg  = g >> 9;                                 // 0: lo half, 1: hi half
    int gl   = g & 511;
    int slot = (gl >> 3) * 16 + (seg << 3) + (gl & 7);

    float* o = coefs + slot * 8;
    o[0] = A * KEXP; o[1] = B * KEXP; o[2] = C * KEXP; o[3] = D * KEXP;
    o[4] = E * KEXP; o[5] = F * KEXP; o[6] = 0.0f;     o[7] = 0.0f;
}

// ---------------------------------------------------------------------------
// Kernel 2: tiled WMMA splat + merge-free front-to-back compositing.
//   Each wave owns 16 pixels (N dim of D).  B operand (pixel features, K=8
//   split across two K=4 f32 WMMAs) is loop-invariant in 4 VGPRs.  Loop over
//   64 gaussian tiles: A operand = 16 permuted gaussians' scaled coefs from
//   LDS (ds_load_2addr_b64).  Lanes 0-15 run the ordered scan over gaussians
//   [0,512), lanes 16-31 over [512,1024); a single xor-16 merge at the end
//   composites the two segments in order.
// ---------------------------------------------------------------------------
__global__ void __launch_bounds__(256)
g2d_splat(const float* __restrict__ coefs, float* __restrict__ out) {
    __shared__ float lds_coefs[NG * 8];               // 32 KB

    // Cooperative stage: 8192 floats = 2048 float4; 256 threads x 8 b128 ops.
    {
        const float4* src = (const float4*)coefs;
        float4*       dst = (float4*)lds_coefs;
        #pragma unroll
        for (int i = 0; i < 8; ++i)
            dst[threadIdx.x + 256 * i] = src[threadIdx.x + 256 * i];
    }
    __syncthreads();

    const int lane = threadIdx.x & 31;
    const int wave = (blockIdx.x * blockDim.x + threadIdx.x) >> 5;
    const int nloc = lane & 15;                        // pixel slot / A row
    const int half = lane >> 4;                        // K-half selector
    const int p    = wave * 16 + nloc;                 // global pixel id

    const float x = -1.0f + (2.0f / 255.0f) * (float)(p & 255);
    const float y = -1.0f + (2.0f / 255.0f) * (float)(p >> 8);

    // B operand (K x N = 8 x 16 over two WMMAs): lanes 0-15 hold K={0,1}/{4,5},
    // lanes 16-31 hold K={2,3}/{6,7}.
    v2f b1, b2;
    if (half == 0) { b1 = (v2f){x * x, x * y}; b2 = (v2f){y, 1.0f}; }
    else           { b1 = (v2f){y * y, x};     b2 = (v2f){0.0f, 0.0f}; }

    float ACC = 0.0f;                                  // segment color
    float T   = 1.0f;                                  // segment transmittance

    for (int t = 0; t < 64; ++t) {
        // A operand row nloc of this tile; lane-half picks K pair.
        const float* cg = lds_coefs + (t * 16 + nloc) * 8 + 2 * half;
        v2f a1 = *(const v2f*)(cg);                    // ds_load_2addr_b64
        v2f a2 = *(const v2f*)(cg + 4);

        v8f q8 = {};
        q8 = __builtin_amdgcn_wmma_f32_16x16x4_f32(false, a1, false, b1,
                                                   (short)0, q8, false, false);
        q8 = __builtin_amdgcn_wmma_f32_16x16x4_f32(false, a2, false, b2,
                                                   (short)0, q8, false, false);

        // Ordered scan over this lane's 8 consecutive segment gaussians,
        // folded directly into the running (ACC, T) carry.  q8[r] <= 0, so
        // bare v_exp_f32 (flush-to-zero underflow) matches the reference.
        #pragma unroll
        for (int r = 0; r < 8; ++r) {
            float al = __builtin_amdgcn_exp2f(q8[r]);  // alpha
            ACC = fmaf(T, al, ACC);
            T   = fmaf(-T, al, T);                     // T *= (1 - alpha)
        }
    }

    // Single ordered merge: segment [0,512) (lanes 0-15, own state) followed
    // by segment [512,1024) (lanes 16-31, fetched via xor-16).
    float oACC = __shfl_xor(ACC, 16, 32);
    if (half == 0) {
        float result = fmaf(T, oACC, ACC);             // bg = 0
        float* o = out + p * 3;
        o[0] = result; o[1] = result; o[2] = result;
    }
}

// ---------------------------------------------------------------------------
extern "C" void kernel_launch(void* const* d_in, const int* in_sizes, int n_in,
                              void* d_out, int out_size, void* d_ws, size_t ws_size,
                              hipStream_t stream) {
    const float* means  = (const float*)d_in[0];
    const float* scales = (const float*)d_in[1];
    const float* thetas = (const float*)d_in[2];
    const float* opac   = (const float*)d_in[3];
    float*       coefs  = (float*)d_ws;               // 1024*8*4 = 32 KB
    float*       out    = (float*)d_out;

    g2d_precompute<<<NG / 256, 256, 0, stream>>>(means, scales, thetas, opac, coefs);
    // 65536 pixels / 16 per wave = 4096 waves; 8 waves per block -> 512 blocks.
    g2d_splat<<<512, 256, 0, stream>>>(coefs, out);
}